// EMMFeatureExtractor_77249281786345
// MI455X (gfx1250) — compile-verified
//
#include <hip/hip_runtime.h>
#include <hip/hip_bf16.h>
#include <stdint.h>

#define ROI_N   512
#define CH      256
#define FH      64
#define FW      64
#define OUT_HW  15
#define SRATE   2
#define CCHUNK  32
#define NCHUNK  (CH / CCHUNK)
#define MAXR    12   // max crop extent per axis (ROI <= ~10 feature px + hi corner)

typedef __attribute__((ext_vector_type(4))) unsigned int uint32x4_;
typedef __attribute__((ext_vector_type(8))) int          int32x8_;
typedef __attribute__((ext_vector_type(4))) int          int32x4_;

#if __has_builtin(__builtin_amdgcn_tensor_load_to_lds)
#define HAVE_TDM 1
#else
#define HAVE_TDM 0
#endif

// Issue one TDM crop load: CCHUNK channel planes of an (Hr x Wr) spatial
// window, packed contiguously into LDS as [c][h][w] (row stride Wr).
__device__ __forceinline__ void tdm_issue(uint64_t gaddr, unsigned lds_off,
                                          int Wr, int Hr) {
#if HAVE_TDM
    // D# group 0: count=1 | lds_addr | global_addr(57b) | type=2
    uint32x4_ g0;
    g0[0] = 1u;
    g0[1] = lds_off;
    g0[2] = (unsigned)(gaddr & 0xFFFFFFFFu);
    g0[3] = ((unsigned)(gaddr >> 32) & 0x01FFFFFFu) | 0x80000000u;
    // D# group 1: data_size=4B, tensor_dim0/1 = 64, tile=(Wr,Hr,CCHUNK),
    // dim0_stride=64 (h step), dim1_stride=4096 (channel step)
    int32x8_ g1;
    g1[0] = 0x00020000;                       // data_size=2 (4 bytes)
    g1[1] = (FW << 16);                       // tensor_dim0 lo16
    g1[2] = (FH << 16);                       // dim0 hi16=0 | tensor_dim1 lo16
    g1[3] = (Wr << 16);                       // dim1 hi16=0 | tile_dim0
    g1[4] = Hr | (CCHUNK << 16);              // tile_dim1 | tile_dim2
    g1[5] = FW;                               // tensor_dim0_stride lo32
    g1[6] = (int)((unsigned)(FH * FW) << 16); // stride0 hi16=0 | stride1 lo16
    g1[7] = 0;                                // tensor_dim1_stride hi32
    // D# group 2/3: 3-D tensor, tile_dim3/4 = 0
    int32x4_ g2; g2[0] = CH; g2[1] = 0; g2[2] = 0; g2[3] = 0;
    int32x4_ g3; g3[0] = 0;  g3[1] = 0; g3[2] = 0; g3[3] = 0;
    // Trailing words (6-arg clang-23 form); count=1 so a zero group is ignored.
    int32x8_ g4; g4[0]=0; g4[1]=0; g4[2]=0; g4[3]=0;
                 g4[4]=0; g4[5]=0; g4[6]=0; g4[7]=0;
    __builtin_amdgcn_tensor_load_to_lds(g0, g1, g2, g3, g4, 0);
#else
    (void)gaddr; (void)lds_off; (void)Wr; (void)Hr;
#endif
}

// One bilinear axis entry: lo/hi offsets (relative to crop origin, scaled by
// stride) and weights, matching the reference _axis_idx semantics.
__device__ __forceinline__ void axis_entry(float p, int size, int origin, int stride,
                                           int* off, float* wt) {
    const float fsize = (float)size;
    const float v  = (p >= -1.0f && p <= fsize) ? 1.0f : 0.0f;
    float pc = fminf(fmaxf(p, 0.0f), fsize - 1.0f);
    int lo = (int)floorf(pc);
    lo = (lo > size - 1) ? (size - 1) : lo;
    int hi = lo + 1;
    hi = (hi > size - 1) ? (size - 1) : hi;
    const float f = pc - (float)lo;
    off[0] = (lo - origin) * stride;
    off[1] = (hi - origin) * stride;
    wt[0]  = (1.0f - f) * v;
    wt[1]  = f * v;
}

__global__ __launch_bounds__(256)
void roi_align_tdm_kernel(const float* __restrict__ feats,
                          const float* __restrict__ boxes,
                          const int*   __restrict__ batch_idx,
                          float*       __restrict__ out)
{
    __shared__ float tile[2][CCHUNK * MAXR * MAXR];   // 2 x 18 KB, double buffer

    const int n = blockIdx.x;
    const int t = threadIdx.x;

    const float scale = 0.0625f;
    const float x1 = boxes[n * 4 + 0] * scale;
    const float y1 = boxes[n * 4 + 1] * scale;
    const float x2 = boxes[n * 4 + 2] * scale;
    const float y2 = boxes[n * 4 + 3] * scale;
    const int   b  = batch_idx[n];

    const float roi_w = fmaxf(x2 - x1, 1.0f);
    const float roi_h = fmaxf(y2 - y1, 1.0f);
    const float bin_w = roi_w * (1.0f / OUT_HW);
    const float bin_h = roi_h * (1.0f / OUT_HW);

    // Crop bounds covering every clamped bilinear corner (sample positions
    // span [start+0.25*bin, start+14.75*bin]).
    const float xs_min = x1 + 0.25f  * bin_w;
    const float xs_max = x1 + 14.75f * bin_w;
    const float ys_min = y1 + 0.25f  * bin_h;
    const float ys_max = y1 + 14.75f * bin_h;
    int w0 = (int)floorf(fminf(fmaxf(xs_min, 0.0f), (float)(FW - 1)));
    int w1 = (int)floorf(fminf(fmaxf(xs_max, 0.0f), (float)(FW - 1))) + 1;
    w1 = (w1 > FW - 1) ? (FW - 1) : w1;
    int h0 = (int)floorf(fminf(fmaxf(ys_min, 0.0f), (float)(FH - 1)));
    int h1 = (int)floorf(fminf(fmaxf(ys_max, 0.0f), (float)(FH - 1))) + 1;
    h1 = (h1 > FH - 1) ? (FH - 1) : h1;

    // Uniform (SGPR) crop descriptors
    const int s_w0 = __builtin_amdgcn_readfirstlane(w0);
    const int s_h0 = __builtin_amdgcn_readfirstlane(h0);
    const int s_Wr = __builtin_amdgcn_readfirstlane(w1 - w0 + 1);
    const int s_Hr = __builtin_amdgcn_readfirstlane(h1 - h0 + 1);
    const int s_b  = __builtin_amdgcn_readfirstlane(b);
    const int HrWr = s_Hr * s_Wr;
    // Low 32 bits of a generic LDS address are the workgroup-relative LDS
    // byte offset.
    const unsigned s_lds0 =
        (unsigned)__builtin_amdgcn_readfirstlane((int)(uint32_t)(uintptr_t)&tile[0][0]);
    const unsigned s_lds1 =
        (unsigned)__builtin_amdgcn_readfirstlane((int)(uint32_t)(uintptr_t)&tile[1][0]);

    // Byte address of the crop origin for channel 0 of this ROI's batch image.
    const uint64_t base_addr = (uint64_t)(uintptr_t)feats +
        4ull * ((uint64_t)(s_b * CH) * (uint64_t)(FH * FW) +
                (uint64_t)s_h0 * FW + (uint64_t)s_w0);

#if HAVE_TDM
    // Pipeline prologue: start chunk 0 DMA; it runs while every thread
    // builds its 16-tap weight table below.
    if (t < 32) tdm_issue(base_addr, s_lds0, s_Wr, s_Hr);
#endif

    // Per-bin 16-tap (offset, weight) table: the 2x2-subsample mean of a
    // separable bilinear kernel is the full 4x4 outer product of per-axis
    // {sub0_lo, sub0_hi, sub1_lo, sub1_hi} entries (x 0.25 at the end).
    int   off16[16];
    float wt16[16];
    if (t < OUT_HW * OUT_HW) {
        const int oy = t / OUT_HW;
        const int ox = t - oy * OUT_HW;
        int   yoffs[4]; float ywts[4];
        int   xoffs[4]; float xwts[4];
#pragma unroll
        for (int s = 0; s < SRATE; ++s) {
            const float ky = ((float)(oy * SRATE + s) + 0.5f) / (float)SRATE;
            const float kx = ((float)(ox * SRATE + s) + 0.5f) / (float)SRATE;
            axis_entry(y1 + ky * bin_h, FH, s_h0, s_Wr, &yoffs[2 * s], &ywts[2 * s]);
            axis_entry(x1 + kx * bin_w, FW, s_w0, 1,    &xoffs[2 * s], &xwts[2 * s]);
        }
#pragma unroll
        for (int a = 0; a < 4; ++a)
#pragma unroll
            for (int c = 0; c < 4; ++c) {
                off16[a * 4 + c] = yoffs[a] + xoffs[c];
                wt16[a * 4 + c]  = ywts[a] * xwts[c];
            }
    }

    for (int i = 0; i < NCHUNK; ++i) {
        const int buf = i & 1;
#if HAVE_TDM
        if (t < 32) {   // wave 0 drives the Tensor Data Mover pipeline
            if (i + 1 < NCHUNK) {
                // Next chunk goes into the other buffer; its previous consumer
                // (chunk i-1) finished at the last end-of-iteration barrier.
                tdm_issue(base_addr + 4ull * (uint64_t)((i + 1) * CCHUNK) *
                                      (uint64_t)(FH * FW),
                          buf ? s_lds0 : s_lds1, s_Wr, s_Hr);
                // TDM ops complete in order: count <= 1 means chunk i landed.
                __builtin_amdgcn_s_wait_tensorcnt(1);
            } else {
                __builtin_amdgcn_s_wait_tensorcnt(0);
            }
        }
#else
        // Fallback: cooperative staged copy (no TDM builtin on this toolchain)
        {
            const int c0 = i * CCHUNK;
            const int total = CCHUNK * HrWr;
            for (int idx = t; idx < total; idx += 256) {
                const int c  = idx / HrWr;
                const int r  = idx - c * HrWr;
                const int hh = r / s_Wr;
                const int ww = r - hh * s_Wr;
                tile[buf][idx] =
                    feats[((size_t)(s_b * CH + c0 + c) * FH + (s_h0 + hh)) * FW
                          + (s_w0 + ww)];
            }
        }
#endif
        __syncthreads();   // chunk i visible to all waves

        if (t < OUT_HW * OUT_HW) {
            // out[n, c, oy, ox]; t == oy*15+ox
            float* op = out + ((size_t)(n * CH + i * CCHUNK)) * (OUT_HW * OUT_HW) + t;
            const float* __restrict__ lp = &tile[buf][0];
            for (int c = 0; c < CCHUNK; ++c) {
                const float* __restrict__ p = lp + c * HrWr;
                float acc = 0.0f;
#pragma unroll
                for (int k = 0; k < 16; ++k)
                    acc = fmaf(wt16[k], p[off16[k]], acc);
                op[(size_t)c * (OUT_HW * OUT_HW)] = 0.25f * acc;
            }
        }
        __syncthreads();   // buf(i) free before iteration i+1 issues chunk i+2
    }
}

extern "C" void kernel_launch(void* const* d_in, const int* in_sizes, int n_in,
                              void* d_out, int out_size, void* d_ws, size_t ws_size,
                              hipStream_t stream) {
    (void)in_sizes; (void)n_in; (void)out_size; (void)d_ws; (void)ws_size;
    const float* x     = (const float*)d_in[0];
    const float* boxes = (const float*)d_in[1];
    const int*   bidx  = (const int*)d_in[2];
    float*       out   = (float*)d_out;
    roi_align_tdm_kernel<<<dim3(ROI_N), dim3(256), 0, stream>>>(x, boxes, bidx, out);
}